// yoloLoss_53884659695984
// MI455X (gfx1250) — compile-verified
//
#include <hip/hip_runtime.h>

// ---------------- problem constants ----------------
#define CH               30                 // B*5 + C = 2*5 + 20
#define CELLS_PER_BLOCK  256
#define TILE_FLOATS      (CELLS_PER_BLOCK * CH)   // 7680 floats per tensor tile
#define TILE_CHUNKS      (TILE_FLOATS / 4)        // 1920 float4 (b128) chunks
#define EPSF             1e-7f

// ---------------- CDNA5 async global->LDS path (guarded) ----------------
#if __has_builtin(__builtin_amdgcn_global_load_async_to_lds_b128) && \
    __has_builtin(__builtin_amdgcn_s_wait_asynccnt)
#define USE_ASYNC_LDS 1
typedef int v4i_vs __attribute__((vector_size(16)));
typedef __attribute__((address_space(1))) v4i_vs as1_v4i;   // global
typedef __attribute__((address_space(3))) v4i_vs as3_v4i;   // LDS
#else
#define USE_ASYNC_LDS 0
#endif

typedef float v2f __attribute__((ext_vector_type(2)));
typedef float v8f __attribute__((ext_vector_type(8)));

__device__ __forceinline__ float iou_xywh(float cx1, float cy1, float w1, float h1,
                                          float cx2, float cy2, float w2, float h2)
{
    float ix1 = fmaxf(cx1 - w1 * 0.5f, cx2 - w2 * 0.5f);
    float iy1 = fmaxf(cy1 - h1 * 0.5f, cy2 - h2 * 0.5f);
    float ix2 = fminf(cx1 + w1 * 0.5f, cx2 + w2 * 0.5f);
    float iy2 = fminf(cy1 + h1 * 0.5f, cy2 + h2 * 0.5f);
    float iw  = fmaxf(ix2 - ix1, 0.0f);
    float ih  = fmaxf(iy2 - iy1, 0.0f);
    float inter = iw * ih;
    float uni   = w1 * h1 + w2 * h2 - inter + EPSF;
    return inter / uni;
}

// Stage 1: per-block partial loss sums. One block == 256 cells.
__global__ __launch_bounds__(CELLS_PER_BLOCK)
void yolo_loss_partial(const float* __restrict__ pred,
                       const float* __restrict__ targ,
                       float* __restrict__ partials)
{
    __shared__ float shp[TILE_FLOATS];   // 30720 B
    __shared__ float sht[TILE_FLOATS];   // 30720 B
    __shared__ float wsum[CELLS_PER_BLOCK / 32];

    const int    tid  = threadIdx.x;
    const size_t base = (size_t)blockIdx.x * TILE_FLOATS;
    const float* gp   = pred + base;     // 16B aligned: 30720 | base bytes
    const float* gt   = targ + base;

#if USE_ASYNC_LDS
    // Coalesced async DMA: lane i moves bytes [16i,16i+16) straight into LDS,
    // bypassing VGPRs; tracked on ASYNCcnt.
    for (int i = tid; i < TILE_CHUNKS; i += CELLS_PER_BLOCK) {
        __builtin_amdgcn_global_load_async_to_lds_b128(
            (as1_v4i*)(gp + 4 * i), (as3_v4i*)(shp + 4 * i), 0, 0);
        __builtin_amdgcn_global_load_async_to_lds_b128(
            (as1_v4i*)(gt + 4 * i), (as3_v4i*)(sht + 4 * i), 0, 0);
    }
    __builtin_amdgcn_s_wait_asynccnt(0);
#else
    for (int i = tid; i < TILE_CHUNKS; i += CELLS_PER_BLOCK) {
        float4 vp = ((const float4*)gp)[i];
        float4 vt = ((const float4*)gt)[i];
        ((float4*)shp)[i] = vp;
        ((float4*)sht)[i] = vt;
    }
#endif
    __syncthreads();

    // ---- per-cell loss (LDS reads are bank-conflict-free: stride 30 words) ----
    const float* P = shp + tid * CH;
    const float* T = sht + tid * CH;

    float p0x = P[0], p0y = P[1], p0w = P[2], p0h = P[3], p0c = P[4];
    float p1x = P[5], p1y = P[6], p1w = P[7], p1h = P[8], p1c = P[9];
    float t0x = T[0], t0y = T[1], t0w = T[2], t0h = T[3], t0c = T[4];
    float t1x = T[5], t1y = T[6], t1w = T[7], t1h = T[8], t1c = T[9];

    // ious vs target box 0 (for responsibility); pair IoUs for coord loss
    float iouA0 = iou_xywh(p0x, p0y, p0w, p0h, t0x, t0y, t0w, t0h); // == iou_pair[0]
    float iouA1 = iou_xywh(p1x, p1y, p1w, p1h, t0x, t0y, t0w, t0h);
    float iouP1 = iou_xywh(p1x, p1y, p1w, p1h, t1x, t1y, t1w, t1h);

    int   best = (iouA1 > iouA0) ? 1 : 0;       // argmax, first-wins on ties
    float obj  = (t0c > 0.0f) ? 1.0f : 0.0f;
    float r0   = (best == 0) ? obj : 0.0f;
    float r1   = (best == 1) ? obj : 0.0f;

    float c0 = p0c - t0c; c0 *= c0;
    float c1 = p1c - t1c; c1 *= c1;
    float lobj   = c0 * r0 + c1 * r1;
    float lnoobj = c0 * (1.0f - r0) + c1 * (1.0f - r1);

    float coord = (1.0f - sqrtf(fmaxf(iouA0, EPSF))) * r0 +
                  (1.0f - sqrtf(fmaxf(iouP1, EPSF))) * r1;

    float cls = 0.0f;
#pragma unroll
    for (int c = 0; c < 20; ++c) {
        float d = P[10 + c] - T[10 + c];
        cls += d * d;
    }
    cls *= obj;

    float v = 5.0f * coord + lobj + 0.5f * lnoobj + cls;

    // ---- wave32 reduction, then 8 waves -> 1 block partial ----
#pragma unroll
    for (int off = 16; off > 0; off >>= 1)
        v += __shfl_down(v, off, 32);
    if ((tid & 31) == 0) wsum[tid >> 5] = v;
    __syncthreads();
    if (tid == 0) {
        float s = 0.0f;
#pragma unroll
        for (int w = 0; w < CELLS_PER_BLOCK / 32; ++w) s += wsum[w];
        partials[blockIdx.x] = s;
    }
}

// Stage 2: single wave32 reduces the partials with V_WMMA_F32_16X16X4_F32.
// A carries 64 partials per step, B = ones => D row m accumulates rowsum(m);
// column n==0 (lanes 0 and 16) holds the 16 row sums.
__global__ __launch_bounds__(32)
void yolo_loss_reduce_wmma(const float* __restrict__ partials,
                           float* __restrict__ out,
                           int n, float invN)
{
    const int lane = threadIdx.x;          // 0..31, EXEC all ones (WMMA req.)
    v8f acc = {};
    v2f ones; ones.x = 1.0f; ones.y = 1.0f;

    const int nFull = n & ~63;
    for (int base = 0; base < nFull; base += 64) {
        v2f a;
        a.x = partials[base + 2 * lane];
        a.y = partials[base + 2 * lane + 1];
        // D = A(16x4 f32) * ones(4x16) + C : accumulates row sums of A
        acc = __builtin_amdgcn_wmma_f32_16x16x4_f32(
            /*neg_a=*/false, a, /*neg_b=*/false, ones,
            /*c_mod=*/(short)0, acc, /*reuse_a=*/false, /*reuse_b=*/false);
    }

    // lane 0 holds rows M=0..7 in acc[0..7], lane 16 holds rows M=8..15 (N==0 col)
    float s = acc[0] + acc[1] + acc[2] + acc[3] + acc[4] + acc[5] + acc[6] + acc[7];
    float sHi = __shfl(s, 16, 32);

    if (lane == 0) {
        float tail = 0.0f;
        for (int i = nFull; i < n; ++i) tail += partials[i];   // n=3136 -> empty
        out[0] = (s + sHi + tail) * invN;
    }
}

extern "C" void kernel_launch(void* const* d_in, const int* in_sizes, int n_in,
                              void* d_out, int out_size, void* d_ws, size_t ws_size,
                              hipStream_t stream)
{
    const float* pred = (const float*)d_in[0];
    const float* targ = (const float*)d_in[1];
    float* out      = (float*)d_out;
    float* partials = (float*)d_ws;

    const int totalElems = in_sizes[0];               // N*S*S*(B*5+C)
    const int nCells     = totalElems / CH;           // N*S*S = 802816
    const int nBlocks    = nCells / CELLS_PER_BLOCK;  // 3136 (exact)
    const int N          = nCells / (14 * 14);        // 4096
    const float invN     = 1.0f / (float)N;

    yolo_loss_partial<<<nBlocks, CELLS_PER_BLOCK, 0, stream>>>(pred, targ, partials);
    yolo_loss_reduce_wmma<<<1, 32, 0, stream>>>(partials, out, nBlocks, invN);
}